// DiffComposition_22514218565909
// MI455X (gfx1250) — compile-verified
//
#include <hip/hip_runtime.h>
#include <cstdint>

#define NLAYER 8
#define CCH    3
#define HIN    512
#define WIN    512
#define HOUT   2048
#define WOUT   2048
#define PXT    8      // pixels per thread (WOUT / 256)
#define EPSV   1e-8f

// ---------------------------------------------------------------------------
// CDNA5 async global->LDS copy (GLOBAL_LOAD_ASYNC_TO_LDS_B128, ASYNCcnt)
// Builtin signature (from clang diagnostic):
//   void __builtin_amdgcn_global_load_async_to_lds_b128(
//       int __attribute__((vector_size(16))) AS(1)* gsrc,
//       int __attribute__((vector_size(16))) AS(3)* lds,
//       int imm_offset, int imm_cpol)
// ---------------------------------------------------------------------------
typedef int v4i __attribute__((vector_size(16)));
typedef v4i __attribute__((address_space(1))) v4i_g;
typedef v4i __attribute__((address_space(3))) v4i_l;
typedef void __attribute__((address_space(3))) lvoid_t;

__device__ __forceinline__ void async_copy_b128(const float* g, float* l) {
#if __has_builtin(__builtin_amdgcn_global_load_async_to_lds_b128)
  __builtin_amdgcn_global_load_async_to_lds_b128((v4i_g*)g, (v4i_l*)l,
                                                 /*offset=*/0, /*cpol=*/0);
#else
  uint32_t lds_off = (uint32_t)(uintptr_t)(lvoid_t*)l;
  asm volatile("global_load_async_to_lds_b128 %0, %1, off"
               :: "v"(lds_off), "v"(g) : "memory");
#endif
}

__device__ __forceinline__ void wait_asynccnt0() {
#if __has_builtin(__builtin_amdgcn_s_wait_asynccnt)
  __builtin_amdgcn_s_wait_asynccnt(0);
#else
  asm volatile("s_wait_asynccnt 0" ::: "memory");
#endif
}

// ---------------------------------------------------------------------------
// Kernel 1: per-layer affine params. ix = cx*x + dx ; iy = cy*y + dy
// Derived from sigmoid box -> theta -> affine_grid -> grid_sample unnormalize
// (align_corners=False), all separable because theta off-diagonals are 0.
// ---------------------------------------------------------------------------
__global__ void DiffComposition_params_kernel(const float* __restrict__ coor,
                                              float* __restrict__ p) {
  const int n = threadIdx.x;
  if (n >= NLAYER) return;
  const float s0 = 1.0f / (1.0f + expf(-coor[4 * n + 0]));
  const float s1 = 1.0f / (1.0f + expf(-coor[4 * n + 1]));
  const float s2 = 1.0f / (1.0f + expf(-coor[4 * n + 2]));
  const float s3 = 1.0f / (1.0f + expf(-coor[4 * n + 3]));
  const float xc = s0 * (float)WOUT;
  const float yc = s1 * (float)HOUT;
  const float w  = s2 * (float)WOUT;
  const float h  = s3 * (float)HOUT;
  const float a  = (float)WOUT / (w + EPSV);
  const float tx = (2.0f / (float)WOUT) * ((float)WOUT * 0.5f - xc) * a;
  const float b  = (float)HOUT / (h + EPSV);
  const float ty = (2.0f / (float)HOUT) * ((float)HOUT * 0.5f - yc) * b;
  // ix = ((a*xs + tx) + 1)*WIN/2 - 0.5 with xs = (2x+1)/WOUT - 1  => linear in x
  const float cx = a * (float)WIN / (float)WOUT;
  const float dx = cx * 0.5f - a * (float)WIN * 0.5f + (tx + 1.0f) * (float)WIN * 0.5f - 0.5f;
  const float cy = b * (float)HIN / (float)HOUT;
  const float dy = cy * 0.5f - b * (float)HIN * 0.5f + (ty + 1.0f) * (float)HIN * 0.5f - 0.5f;
  p[4 * n + 0] = cx;
  p[4 * n + 1] = dx;
  p[4 * n + 2] = cy;
  p[4 * n + 3] = dy;
}

// ---------------------------------------------------------------------------
// Kernel 2: one block per output row; composite all 8 layers in registers.
// Per layer: stage the 2 needed source rows x 3 channels (12KB) into LDS via
// async copies, then separable bilinear gather from LDS.
// ---------------------------------------------------------------------------
__global__ __launch_bounds__(256) void DiffComposition_composite_kernel(
    const float* __restrict__ src,   // [8,3,512,512]
    const float* __restrict__ bg,    // [1,3,2048,2048]
    const float* __restrict__ prm,   // [8,4] = cx,dx,cy,dy
    float* __restrict__ out) {       // [1,3,2048,2048]
  __shared__ __align__(16) float rowbuf[2 * CCH][WIN];  // 12 KB

  const int y = blockIdx.x;
  const int t = threadIdx.x;

  float acc[CCH][PXT];
#pragma unroll
  for (int c = 0; c < CCH; ++c)
#pragma unroll
    for (int k = 0; k < PXT; ++k)
      acc[c][k] = __builtin_nontemporal_load(&bg[((size_t)c * HOUT + y) * WOUT + (t + 256 * k)]);

  for (int n = 0; n < NLAYER; ++n) {
    const float cx = prm[4 * n + 0];
    const float dx = prm[4 * n + 1];
    const float cy = prm[4 * n + 2];
    const float dy = prm[4 * n + 3];

    const float iy  = cy * (float)y + dy;
    const float y0f = floorf(iy);
    const float wy1 = iy - y0f;
    const float wy0 = 1.0f - wy1;
    const int   y0  = (int)y0f;
    const int   y1  = y0 + 1;
    const bool  v0  = (y0 >= 0) && (y0 < HIN);
    const bool  v1  = (y1 >= 0) && (y1 < HIN);
    if (!v0 && !v1) continue;  // uniform across the block: whole layer misses this row
    const float wyv0 = v0 ? wy0 : 0.0f;
    const float wyv1 = v1 ? wy1 : 0.0f;
    const float wys  = wyv0 + wyv1;
    const int   y0c  = min(max(y0, 0), HIN - 1);
    const int   y1c  = min(max(y1, 0), HIN - 1);

    __syncthreads();  // previous layer's LDS reads are done before we overwrite
#pragma unroll
    for (int j = 0; j < 3; ++j) {          // 768 x b128 chunks = 6 rows * 2KB
      const int idx  = t + 256 * j;        // chunk id 0..767
      const int r    = idx >> 7;           // LDS row 0..5 (128 chunks per row)
      const int col  = (idx & 127) << 2;   // float column
      const int ch   = r >> 1;
      const int srow = (r & 1) ? y1c : y0c;
      async_copy_b128(&src[((((size_t)n * CCH + ch) * HIN) + srow) * WIN + col],
                      &rowbuf[r][col]);
    }
    wait_asynccnt0();
    __syncthreads();  // all waves' staged rows visible

#pragma unroll
    for (int k = 0; k < PXT; ++k) {
      const int   x   = t + 256 * k;
      const float ix  = cx * (float)x + dx;
      const float x0f = floorf(ix);
      const float wx1 = ix - x0f;
      const float wx0 = 1.0f - wx1;
      const int   x0  = (int)x0f;
      const int   x1  = x0 + 1;
      const float wxv0 = ((x0 >= 0) && (x0 < WIN)) ? wx0 : 0.0f;
      const float wxv1 = ((x1 >= 0) && (x1 < WIN)) ? wx1 : 0.0f;
      const float wxs  = wxv0 + wxv1;
      if (wxs > 0.0f) {  // out-of-box pixels: m == 0, blend is a no-op
        const int   x0c = min(max(x0, 0), WIN - 1);
        const int   x1c = min(max(x1, 0), WIN - 1);
        const float m   = wys * wxs;
        const float om  = 1.0f - m;
#pragma unroll
        for (int c = 0; c < CCH; ++c) {
          const float s =
              (rowbuf[2 * c][x0c] * wyv0 + rowbuf[2 * c + 1][x0c] * wyv1) * wxv0 +
              (rowbuf[2 * c][x1c] * wyv0 + rowbuf[2 * c + 1][x1c] * wyv1) * wxv1;
          acc[c][k] = acc[c][k] * om + s * m;
        }
      }
    }
  }

#pragma unroll
  for (int c = 0; c < CCH; ++c)
#pragma unroll
    for (int k = 0; k < PXT; ++k)
      __builtin_nontemporal_store(acc[c][k],
                                  &out[((size_t)c * HOUT + y) * WOUT + (t + 256 * k)]);
}

// ---------------------------------------------------------------------------
extern "C" void kernel_launch(void* const* d_in, const int* in_sizes, int n_in,
                              void* d_out, int out_size, void* d_ws, size_t ws_size,
                              hipStream_t stream) {
  (void)in_sizes; (void)n_in; (void)out_size; (void)ws_size;
  const float* src  = (const float*)d_in[0];  // (8,3,512,512)
  const float* bg   = (const float*)d_in[1];  // (1,3,2048,2048)
  const float* coor = (const float*)d_in[2];  // (8,4)
  float* out = (float*)d_out;
  float* prm = (float*)d_ws;                  // 32 floats

  DiffComposition_params_kernel<<<1, 32, 0, stream>>>(coor, prm);
  DiffComposition_composite_kernel<<<HOUT, 256, 0, stream>>>(src, bg, prm, out);
}